// SimilarityModel_36885179138523
// MI455X (gfx1250) — compile-verified
//
#include <hip/hip_runtime.h>

#define HIDDEN 128
#define BSZ    8
#define CDIM   512
#define QDIM   64
#define K3     384   // 3*HIDDEN

typedef float v2f __attribute__((ext_vector_type(2)));
typedef float v8f __attribute__((ext_vector_type(8)));

__device__ __forceinline__ v8f wmma_f32(v2f a, v2f b, v8f c) {
  // V_WMMA_F32_16X16X4_F32 : D = A(16x4) * B(4x16) + C(16x16), all fp32
  return __builtin_amdgcn_wmma_f32_16x16x4_f32(
      /*neg_a=*/false, a, /*neg_b=*/false, b,
      /*c_mod=*/(short)0, c, /*reuse_a=*/false, /*reuse_b=*/false);
}

__global__ __launch_bounds__(256)
void SimilarityModel_fused_kernel(const float* __restrict__ hg,
                                  const float* __restrict__ ug,
                                  const float* __restrict__ w1,
                                  const float* __restrict__ b1,
                                  const float* __restrict__ w2,
                                  const float* __restrict__ b2,
                                  float* __restrict__ out)
{
  // pitch 132 (=128+4) so 16 rows hit 16 distinct LDS banks
  __shared__ __align__(16) float h_s[16][132];
  __shared__ __align__(16) float u_s[16][132];
  __shared__ float b1_s[HIDDEN];
  __shared__ float w2_s[HIDDEN];

  const int c_tile = blockIdx.x;   // 0..31
  const int q_tile = blockIdx.y;   // 0..3
  const int b      = blockIdx.z;   // 0..7

  const int tid    = threadIdx.x;  // 0..255 (8 wave32)
  const int wave   = tid >> 5;
  const int lane   = tid & 31;
  const int lane16 = lane & 15;
  const int hi     = (lane >> 4) & 1;

  const float* hgt = hg + (size_t)(b * CDIM + c_tile * 16) * HIDDEN;
  const float* ugt = ug + (size_t)(b * QDIM + q_tile * 16) * HIDDEN;

  // ---- stage 16x128 h and u tiles into LDS (float4 per thread x2) ----
  #pragma unroll
  for (int t = 0; t < 2; ++t) {
    int idx = tid + t * 256;        // 0..511 ; 32 float4 per row
    int row = idx >> 5;
    int c4  = idx & 31;
    float4 hv = ((const float4*)(hgt + row * HIDDEN))[c4];
    float4 uv = ((const float4*)(ugt + row * HIDDEN))[c4];
    *((float4*)&h_s[row][c4 * 4]) = hv;
    *((float4*)&u_s[row][c4 * 4]) = uv;
  }
  if (tid < HIDDEN) {
    b1_s[tid] = b1[tid];
    w2_s[tid] = w2[tid];
  }
  __syncthreads();

  const int c0 = wave;       // this wave's two c-rows of the tile
  const int c1 = wave + 8;

  float p0[8], p1[8];        // per-q partial sums of w2[k]*relu(l1)
  #pragma unroll
  for (int v = 0; v < 8; ++v) { p0[v] = 0.f; p1[v] = 0.f; }

  for (int nt = 0; nt < 8; ++nt) {   // 16-wide k (output-channel) tiles
    // B fragment source: w1 row k = nt*16+lane16, K-rows 2*hi+{0,1}
    const float* wrow = w1 + (size_t)(nt * 16 + lane16) * K3 + 2 * hi;

    v8f acc0 = {0.f,0.f,0.f,0.f,0.f,0.f,0.f,0.f};
    v8f acc1 = {0.f,0.f,0.f,0.f,0.f,0.f,0.f,0.f};
    v8f accU = {0.f,0.f,0.f,0.f,0.f,0.f,0.f,0.f};

    // segment 1: j in [0,128)  A[m][j] = h[c][j]  (broadcast over q)
    #pragma unroll 4
    for (int kk = 0; kk < 32; ++kk) {
      int j  = kk * 4 + 2 * hi;
      v2f bf = *(const v2f*)(wrow + kk * 4);
      v2f a0 = *(const v2f*)&h_s[c0][j];
      v2f a1 = *(const v2f*)&h_s[c1][j];
      acc0 = wmma_f32(a0, bf, acc0);
      acc1 = wmma_f32(a1, bf, acc1);
    }
    // segment 2: j in [128,256)  A[m][j] = u[q][j-128]  (c-independent -> shared)
    #pragma unroll 4
    for (int kk = 0; kk < 32; ++kk) {
      int j  = kk * 4 + 2 * hi;
      v2f bf = *(const v2f*)(wrow + 128 + kk * 4);
      v2f au = *(const v2f*)&u_s[lane16][j];
      accU = wmma_f32(au, bf, accU);
    }
    // segment 3: j in [256,384)  A[m][j] = h[c][j-256]*u[q][j-256]
    #pragma unroll 4
    for (int kk = 0; kk < 32; ++kk) {
      int j  = kk * 4 + 2 * hi;
      v2f bf = *(const v2f*)(wrow + 256 + kk * 4);
      v2f au = *(const v2f*)&u_s[lane16][j];
      v2f h0 = *(const v2f*)&h_s[c0][j];
      v2f h1 = *(const v2f*)&h_s[c1][j];
      v2f a0 = h0 * au;
      v2f a1 = h1 * au;
      acc0 = wmma_f32(a0, bf, acc0);
      acc1 = wmma_f32(a1, bf, acc1);
    }

    // epilogue for this k-tile: bias, relu, scale by w2, accumulate
    const int   k   = nt * 16 + lane16;
    const float b1k = b1_s[k];
    const float w2k = w2_s[k];
    #pragma unroll
    for (int v = 0; v < 8; ++v) {
      float t0 = fmaxf(acc0[v] + accU[v] + b1k, 0.f);
      float t1 = fmaxf(acc1[v] + accU[v] + b1k, 0.f);
      p0[v] = fmaf(w2k, t0, p0[v]);
      p1[v] = fmaf(w2k, t1, p1[v]);
    }
  }

  // k-reduction: butterfly across the 16 lanes of each half-wave
  #pragma unroll
  for (int m = 8; m >= 1; m >>= 1) {
    #pragma unroll
    for (int v = 0; v < 8; ++v) {
      p0[v] += __shfl_xor(p0[v], m, 32);
      p1[v] += __shfl_xor(p1[v], m, 32);
    }
  }

  const float b2v = b2[0];
  if (lane16 == 0) {
    // lane 0 holds q rows 0..7 (v=0..7), lane 16 holds q rows 8..15
    size_t base = ((size_t)(b * CDIM + c_tile * 16)) * QDIM
                + (size_t)q_tile * 16 + 8 * hi;
    float* o0 = out + base + (size_t)c0 * QDIM;
    float* o1 = out + base + (size_t)c1 * QDIM;
    #pragma unroll
    for (int v = 0; v < 8; ++v) {
      o0[v] = fmaxf(p0[v] + b2v, 0.f);
      o1[v] = fmaxf(p1[v] + b2v, 0.f);
    }
  }
}

extern "C" void kernel_launch(void* const* d_in, const int* in_sizes, int n_in,
                              void* d_out, int out_size, void* d_ws, size_t ws_size,
                              hipStream_t stream) {
  const float* h  = (const float*)d_in[0];
  const float* u  = (const float*)d_in[1];
  const float* w1 = (const float*)d_in[2];
  const float* b1 = (const float*)d_in[3];
  const float* w2 = (const float*)d_in[4];
  const float* b2 = (const float*)d_in[5];
  float* out = (float*)d_out;

  dim3 grid(CDIM / 16, QDIM / 16, BSZ);   // 32 x 4 x 8 = 1024 tiles
  SimilarityModel_fused_kernel<<<grid, 256, 0, stream>>>(h, u, w1, b1, w2, b2, out);
}